// HitTheMiddleModel_56538949485038
// MI455X (gfx1250) — compile-verified
//
#include <hip/hip_runtime.h>

typedef float v4f __attribute__((ext_vector_type(4)));

// Per-row physics + tiny linear, matching the JAX reference op-for-op.
__device__ __forceinline__ void sim_row(
    float x0, float x1, float x2,
    float w00, float w01, float w02,
    float w10, float w11, float w12,
    float w20, float w21, float w22,
    float b0, float b1, float b2,
    float& o0, float& o1, float& o2)
{
    float vel = x1 + x2;
    float pos = x0 + vel;
    bool hit = (pos > 10.0f) || (pos < -10.0f);
    vel = hit ? -vel : vel;
    pos = fminf(10.0f, fmaxf(-10.0f, pos));
    float reward = -(pos * pos);
    // y = x @ W.T + b  (rows of W in SGPRs)
    float y0 = fmaf(w00, x0, fmaf(w01, x1, fmaf(w02, x2, b0)));
    float y1 = fmaf(w10, x0, fmaf(w11, x1, fmaf(w12, x2, b1)));
    float y2 = fmaf(w20, x0, fmaf(w21, x1, fmaf(w22, x2, b2)));
    o0 = fmaf(1e-10f, y0, pos);
    o1 = fmaf(1e-10f, y1, vel);
    o2 = fmaf(1e-10f, y2, reward);
}

// Each thread processes 4 rows = 12 floats = 3x b128 load + 3x b128 NT store.
__global__ __launch_bounds__(256) void hitmid_vec_kernel(
    const float* __restrict__ x,
    const float* __restrict__ W,
    const float* __restrict__ b,
    float* __restrict__ out,
    int groups)   // number of 4-row groups
{
    int t = blockIdx.x * blockDim.x + threadIdx.x;
    if (t >= groups) return;

    // Uniform address + constant offsets -> scalar loads into SGPRs.
    float w00 = W[0], w01 = W[1], w02 = W[2];
    float w10 = W[3], w11 = W[4], w12 = W[5];
    float w20 = W[6], w21 = W[7], w22 = W[8];
    float b0 = b[0], b1 = b[1], b2 = b[2];

    const size_t base = (size_t)t * 3;           // in float4 units
    const v4f* __restrict__ xv = (const v4f*)x + base;

    // gfx1250 stream-ahead prefetch (~3 MB ahead); OOB speculative prefetch
    // is silently dropped by hardware.
    __builtin_prefetch((const char*)(xv + 3 * 65536), 0, 1);

    v4f f0 = xv[0];
    v4f f1 = xv[1];
    v4f f2 = xv[2];

    // Unpack 4 rows: r0={f0.x,f0.y,f0.z} r1={f0.w,f1.x,f1.y}
    //                r2={f1.z,f1.w,f2.x} r3={f2.y,f2.z,f2.w}
    float a0, a1, a2, c0, c1, c2, d0, d1, d2, e0, e1, e2;
    sim_row(f0.x, f0.y, f0.z, w00,w01,w02, w10,w11,w12, w20,w21,w22, b0,b1,b2,
            a0, a1, a2);
    sim_row(f0.w, f1.x, f1.y, w00,w01,w02, w10,w11,w12, w20,w21,w22, b0,b1,b2,
            c0, c1, c2);
    sim_row(f1.z, f1.w, f2.x, w00,w01,w02, w10,w11,w12, w20,w21,w22, b0,b1,b2,
            d0, d1, d2);
    sim_row(f2.y, f2.z, f2.w, w00,w01,w02, w10,w11,w12, w20,w21,w22, b0,b1,b2,
            e0, e1, e2);

    v4f g0, g1, g2;
    g0.x = a0; g0.y = a1; g0.z = a2; g0.w = c0;
    g1.x = c1; g1.y = c2; g1.z = d0; g1.w = d1;
    g2.x = d2; g2.y = e0; g2.z = e1; g2.w = e2;

    // Non-temporal stores: rinse the write stream out of L2 early so the
    // 100 MB input stays resident in the 192 MB L2 across graph replays.
    v4f* __restrict__ ov = (v4f*)out + base;
    __builtin_nontemporal_store(g0, ov + 0);
    __builtin_nontemporal_store(g1, ov + 1);
    __builtin_nontemporal_store(g2, ov + 2);
}

// Scalar tail for row counts not divisible by 4 (not hit for B=8388608,
// but keeps the kernel correct for any size).
__global__ void hitmid_tail_kernel(
    const float* __restrict__ x,
    const float* __restrict__ W,
    const float* __restrict__ b,
    float* __restrict__ out,
    int start_row, int rows)
{
    int r = start_row + blockIdx.x * blockDim.x + threadIdx.x;
    if (r >= rows) return;
    float w00 = W[0], w01 = W[1], w02 = W[2];
    float w10 = W[3], w11 = W[4], w12 = W[5];
    float w20 = W[6], w21 = W[7], w22 = W[8];
    float b0 = b[0], b1 = b[1], b2 = b[2];
    size_t k = (size_t)r * 3;
    float o0, o1, o2;
    sim_row(x[k], x[k+1], x[k+2], w00,w01,w02, w10,w11,w12, w20,w21,w22,
            b0,b1,b2, o0, o1, o2);
    out[k]   = o0;
    out[k+1] = o1;
    out[k+2] = o2;
}

extern "C" void kernel_launch(void* const* d_in, const int* in_sizes, int n_in,
                              void* d_out, int out_size, void* d_ws, size_t ws_size,
                              hipStream_t stream) {
    const float* x = (const float*)d_in[0];
    const float* W = (const float*)d_in[1];
    const float* b = (const float*)d_in[2];
    float* out = (float*)d_out;

    long rows = (long)in_sizes[0] / 3;
    int groups = (int)(rows / 4);
    int tail_start = groups * 4;
    int tail = (int)(rows - tail_start);

    if (groups > 0) {
        const int threads = 256;                       // 8 wave32 waves
        int blocks = (groups + threads - 1) / threads; // 8192 for B=8.4M
        hitmid_vec_kernel<<<blocks, threads, 0, stream>>>(x, W, b, out, groups);
    }
    if (tail > 0) {
        hitmid_tail_kernel<<<1, 32, 0, stream>>>(x, W, b, out, tail_start, (int)rows);
    }
}